// GodAreaModel_5729486373550
// MI455X (gfx1250) — compile-verified
//
#include <hip/hip_runtime.h>

typedef float v2f __attribute__((ext_vector_type(2)));
typedef float v4f __attribute__((ext_vector_type(4)));
typedef float v8f __attribute__((ext_vector_type(8)));

constexpr int BSZ = 64;      // batch
constexpr int DIN = 2048;    // input dim
constexpr int NN  = 8192;    // neurons
constexpr int NA  = 4;       // areas
constexpr int AS  = 2048;    // area size
constexpr int NT  = 4;       // ticks
constexpr int VOC = 8192;    // vocab
constexpr float THR = 0.05f;

__device__ __forceinline__ v8f wmma4(v2f a, v2f b, v8f c) {
  // D = A(16x4,f32) x B(4x16,f32) + C(16x16,f32)
  return __builtin_amdgcn_wmma_f32_16x16x4_f32(false, a, false, b, (short)0, c,
                                               false, false);
}
#define LO2(v) __builtin_shufflevector(v, v, 0, 1)
#define HI2(v) __builtin_shufflevector(v, v, 2, 3)

// Write one wave's 64x16 partial tile into its LDS slot.
__device__ __forceinline__ void stash_tile(float* slot, const v8f acc[4],
                                           int lane) {
  const int n16 = lane & 15;
  const int rofs = (lane >> 4) << 3;
#pragma unroll
  for (int mt = 0; mt < 4; ++mt)
#pragma unroll
    for (int j = 0; j < 8; ++j)
      slot[(mt * 16 + rofs + j) * 16 + n16] = acc[mt][j];
}

// ---------------- Receptor stage: Z0[a,b,s] = tanh(x @ rw[idx]^T + rb[idx]) --
__global__ __launch_bounds__(256) void k_receptor(
    const float* __restrict__ x, const int* __restrict__ area_idx,
    const float* __restrict__ rw, const float* __restrict__ rb,
    float* __restrict__ z0 /* [NA][BSZ][AS] */) {
  __shared__ float lds[8 * 1024];
  const int tid = threadIdx.x, wave = tid >> 5, lane = tid & 31;
  const int a = blockIdx.x >> 7, ntile = blockIdx.x & 127;
  const int n16 = lane & 15, h4 = (lane >> 4) << 2;  // 0 or +4 floats
  const int neuron = area_idx[a * AS + ntile * 16 + n16];
  const float* __restrict__ brow = rw + (size_t)neuron * DIN + h4;
  const int kbeg = wave * (DIN / 8);

  v8f acc[4] = {};
#pragma unroll 2
  for (int k0 = kbeg; k0 < kbeg + DIN / 8; k0 += 8) {
    v4f bv = *(const v4f*)(brow + k0);
#pragma unroll
    for (int mt = 0; mt < 4; ++mt) {
      v4f av = *(const v4f*)(x + (size_t)(mt * 16 + n16) * DIN + h4 + k0);
      acc[mt] = wmma4(LO2(av), LO2(bv), acc[mt]);
      acc[mt] = wmma4(HI2(av), HI2(bv), acc[mt]);
    }
  }
  stash_tile(lds + wave * 1024, acc, lane);
  __syncthreads();

  v4f s = {};
#pragma unroll
  for (int w = 0; w < 8; ++w) s += *(const v4f*)(lds + w * 1024 + tid * 4);
  const int row = tid >> 2, c0 = (tid & 3) << 2;
  v4f res;
#pragma unroll
  for (int q = 0; q < 4; ++q) {
    float bb = rb[area_idx[a * AS + ntile * 16 + c0 + q]];
    res[q] = tanhf(s[q] + bb);
  }
  *(v4f*)(z0 + (size_t)a * BSZ * AS + (size_t)row * AS + ntile * 16 + c0) = res;
}

// ---------------- Gate: gate[a] = mean(|Z[a]|) > THR ------------------------
__global__ __launch_bounds__(256) void k_gate(const float* __restrict__ z,
                                              float* __restrict__ gate) {
  __shared__ float red[256];
  const float* __restrict__ p = z + (size_t)blockIdx.x * BSZ * AS;
  float s = 0.f;
  for (int i = threadIdx.x; i < BSZ * AS; i += 256) s += fabsf(p[i]);
  red[threadIdx.x] = s;
  __syncthreads();
  for (int o = 128; o > 0; o >>= 1) {
    if (threadIdx.x < o) red[threadIdx.x] += red[threadIdx.x + o];
    __syncthreads();
  }
  if (threadIdx.x == 0)
    gate[blockIdx.x] = (red[0] * (1.0f / (BSZ * AS)) > THR) ? 1.0f : 0.0f;
}

// ---------------- Weff precompute: weff = W * clip(mask,0,1) ----------------
__global__ __launch_bounds__(256) void k_weff(const float* __restrict__ W,
                                              const float* __restrict__ mask,
                                              float* __restrict__ weff) {
  const size_t g = ((size_t)blockIdx.x * 256 + threadIdx.x) * 4;
  v4f wv = *(const v4f*)(W + g);
  v4f mv = *(const v4f*)(mask + g);
  v4f r;
  r.x = wv.x * fminf(fmaxf(mv.x, 0.f), 1.f);
  r.y = wv.y * fminf(fmaxf(mv.y, 0.f), 1.f);
  r.z = wv.z * fminf(fmaxf(mv.z, 0.f), 1.f);
  r.w = wv.w * fminf(fmaxf(mv.w, 0.f), 1.f);
  *(v4f*)(weff + g) = r;
}

// ---------------- Tick: Z'[o,b,t] = tanh(sum_i g_i (Z[i] @ Weff[o,i]^T) + g.b)
// Block = 8 waves: wave w handles input-area i=w>>1, K-half w&1.
// FUSED=true: Wsrc=W, apply clip(mask) inline. FUSED=false: Wsrc=Weff.
template <bool FUSED>
__global__ __launch_bounds__(256) void k_tick(
    const float* __restrict__ zin, const float* __restrict__ Wsrc,
    const float* __restrict__ mask, const float* __restrict__ bias,
    const float* __restrict__ gate, float* __restrict__ zout) {
  __shared__ float lds[8 * 1024];
  const int tid = threadIdx.x, wave = tid >> 5, lane = tid & 31;
  const int o = blockIdx.x >> 7, ntile = blockIdx.x & 127;
  const int n16 = lane & 15, h4 = (lane >> 4) << 2;
  const int tglob = ntile * 16 + n16;
  const int i = wave >> 1;
  const int kbeg = (wave & 1) * (AS / 2);

  v8f acc[4] = {};
  if (gate[i] > 0.5f) {  // wave-uniform branch: EXEC all-ones at the WMMAs
    const size_t blk = ((size_t)o * NA + i) * (size_t)AS * AS;
    const float* __restrict__ wrow = Wsrc + blk + (size_t)tglob * AS + h4;
    const float* __restrict__ mrow = mask + blk + (size_t)tglob * AS + h4;
    const float* __restrict__ zrow = zin + (size_t)i * BSZ * AS + h4;
#pragma unroll 2
    for (int k0 = kbeg; k0 < kbeg + AS / 2; k0 += 8) {
      v4f bv = *(const v4f*)(wrow + k0);
      if (FUSED) {
        v4f mv = *(const v4f*)(mrow + k0);
        bv.x *= fminf(fmaxf(mv.x, 0.f), 1.f);
        bv.y *= fminf(fmaxf(mv.y, 0.f), 1.f);
        bv.z *= fminf(fmaxf(mv.z, 0.f), 1.f);
        bv.w *= fminf(fmaxf(mv.w, 0.f), 1.f);
      }
#pragma unroll
      for (int mt = 0; mt < 4; ++mt) {
        v4f av = *(const v4f*)(zrow + (size_t)(mt * 16 + n16) * AS + k0);
        acc[mt] = wmma4(LO2(av), LO2(bv), acc[mt]);
        acc[mt] = wmma4(HI2(av), HI2(bv), acc[mt]);
      }
    }
  }
  stash_tile(lds + wave * 1024, acc, lane);  // gate-off waves stash zeros
  __syncthreads();

  v4f s = {};
#pragma unroll
  for (int w = 0; w < 8; ++w) s += *(const v4f*)(lds + w * 1024 + tid * 4);
  const int row = tid >> 2, c0 = (tid & 3) << 2;
  v4f res;
#pragma unroll
  for (int q = 0; q < 4; ++q) {
    const int cg = ntile * 16 + c0 + q;
    float bt = 0.f;
#pragma unroll
    for (int i2 = 0; i2 < NA; ++i2)
      bt += gate[i2] * bias[((size_t)o * NA + i2) * AS + cg];
    res[q] = tanhf(s[q] + bt);
  }
  *(v4f*)(zout + (size_t)o * BSZ * AS + (size_t)row * AS + ntile * 16 + c0) = res;
}

// ---------------- Scatter areas back into flat neuron order -----------------
__global__ __launch_bounds__(256) void k_scatter(
    const float* __restrict__ z, const int* __restrict__ area_idx,
    float* __restrict__ zflat /* [BSZ][NN] */) {
  const int g = blockIdx.x * 256 + threadIdx.x;  // over BSZ*NN
  const int b = g >> 13, j = g & (NN - 1);
  zflat[(size_t)b * NN + area_idx[j]] =
      z[(size_t)(j >> 11) * BSZ * AS + (size_t)b * AS + (j & (AS - 1))];
}

// ---------------- Output: logits = Z_flat @ output_w^T + output_b -----------
__global__ __launch_bounds__(256) void k_output(
    const float* __restrict__ zflat, const float* __restrict__ ow,
    const float* __restrict__ ob, float* __restrict__ logits) {
  __shared__ float lds[8 * 1024];
  const int tid = threadIdx.x, wave = tid >> 5, lane = tid & 31;
  const int ntile = blockIdx.x;  // 0..511
  const int n16 = lane & 15, h4 = (lane >> 4) << 2;
  const int v = ntile * 16 + n16;
  const float* __restrict__ brow = ow + (size_t)v * NN + h4;
  const int kbeg = wave * (NN / 8);

  v8f acc[4] = {};
#pragma unroll 2
  for (int k0 = kbeg; k0 < kbeg + NN / 8; k0 += 8) {
    v4f bv = *(const v4f*)(brow + k0);
#pragma unroll
    for (int mt = 0; mt < 4; ++mt) {
      v4f av = *(const v4f*)(zflat + (size_t)(mt * 16 + n16) * NN + h4 + k0);
      acc[mt] = wmma4(LO2(av), LO2(bv), acc[mt]);
      acc[mt] = wmma4(HI2(av), HI2(bv), acc[mt]);
    }
  }
  stash_tile(lds + wave * 1024, acc, lane);
  __syncthreads();

  v4f s = {};
#pragma unroll
  for (int w = 0; w < 8; ++w) s += *(const v4f*)(lds + w * 1024 + tid * 4);
  const int row = tid >> 2, c0 = (tid & 3) << 2;
  v4f res;
#pragma unroll
  for (int q = 0; q < 4; ++q) res[q] = s[q] + ob[ntile * 16 + c0 + q];
  *(v4f*)(logits + (size_t)row * VOC + ntile * 16 + c0) = res;
}

extern "C" void kernel_launch(void* const* d_in, const int* in_sizes, int n_in,
                              void* d_out, int out_size, void* d_ws, size_t ws_size,
                              hipStream_t stream) {
  const float* x        = (const float*)d_in[0];
  const int*   area_idx = (const int*)d_in[1];
  const float* rw       = (const float*)d_in[2];
  const float* rb       = (const float*)d_in[3];
  const float* W        = (const float*)d_in[4];
  const float* bias     = (const float*)d_in[5];
  const float* mask     = (const float*)d_in[6];
  const float* ow       = (const float*)d_in[7];
  const float* ob       = (const float*)d_in[8];

  float* logits = (float*)d_out;                  // [BSZ][VOC]
  float* hist   = logits + (size_t)BSZ * VOC;     // [NT+1][NA][BSZ][AS]
  float* gate   = (float*)d_ws;                   // [NA]
  float* big    = gate + 32;                      // Weff (256 MB) or Zflat (2 MB)

  const size_t slab = (size_t)NA * BSZ * AS;
  const size_t weff_elems = (size_t)NA * NA * AS * AS;  // 64M floats
  // Zflat (used only after the last tick) aliases the Weff region.
  const bool use_weff = ws_size >= (32 + weff_elems) * sizeof(float);

  // hist[0] = receptor tanh
  k_receptor<<<512, 256, 0, stream>>>(x, area_idx, rw, rb, hist);

  // Optional: fold mask into Weff once (saves one full mask stream per tick).
  if (use_weff)
    k_weff<<<(unsigned)(weff_elems / (256 * 4)), 256, 0, stream>>>(W, mask, big);

  // ticks 1..NT: gate reduce then gated GEMM, ping-pong through hist slabs
  for (int t = 1; t <= NT; ++t) {
    const float* zin = hist + (size_t)(t - 1) * slab;
    float*       zo  = hist + (size_t)t * slab;
    k_gate<<<NA, 256, 0, stream>>>(zin, gate);
    if (use_weff)
      k_tick<false><<<512, 256, 0, stream>>>(zin, big, mask, bias, gate, zo);
    else
      k_tick<true><<<512, 256, 0, stream>>>(zin, W, mask, bias, gate, zo);
  }

  // scatter final Z into flat neuron order, then output projection
  k_scatter<<<(BSZ * NN) / 256, 256, 0, stream>>>(hist + (size_t)NT * slab,
                                                  area_idx, big);
  k_output<<<512, 256, 0, stream>>>(big, ow, ob, logits);
}